// GraphSage_23063974380116
// MI455X (gfx1250) — compile-verified
//
#include <hip/hip_runtime.h>

// GraphSAGE 'pool' (2 layers) for MI455X / gfx1250, wave32.
// N=100000 nodes, E=1600000 edges, D_IN=D_HID=128, D_OUT=40, all fp32.
//
// Roofline: GEMMs are ~15 GFLOP total (trivial); edge gather/scatter-max is
// the bottleneck and is L2-resident (hp = 51.2MB << 192MB L2). GEMMs use
// exact-f32 WMMA (v_wmma_f32_16x16x4_f32). segment_max of relu'd (>=0)
// values == atomicMax on the uint bit pattern with 0-init, which also
// reproduces the "zero-in-degree -> 0" rule exactly.

#define NODES 100000
#define EDGES 1600000
#define DFEAT 128

typedef float v2f __attribute__((ext_vector_type(2)));
typedef float v8f __attribute__((ext_vector_type(8)));

// ---------------------------------------------------------------------------
// WMMA GEMM: out[N, NC] = act( X[N,128] @ W1[128,NC] (+ G[N,128] @ W2[128,NC]) + bias )
// One block = 16 rows. One wave = one 16-col tile (NC=128 -> 8 waves,
// NC=40 -> 3 waves with masked tail). K=128 -> 32 wmma_f32_16x16x4 per GEMM.
// ---------------------------------------------------------------------------
template <int NC, bool FUSED, bool RELU>
__global__ __launch_bounds__(256) void sage_gemm(
    const float* __restrict__ X, const float* __restrict__ G,
    const float* __restrict__ W1, const float* __restrict__ W2,
    const float* __restrict__ bias, float* __restrict__ out)
{
    // pitch 132: 132 % 64 == 4 -> conflict-free column reads of the A slab
    __shared__ float As[16][132];
    __shared__ float Gs[16][132];

    const int  tid     = threadIdx.x;
    const int  nthr    = blockDim.x;
    const long rowBase = (long)blockIdx.x * 16;

    // Cooperative stage of the 16x128 row slab(s) into LDS (float4 loads).
    for (int i = tid; i < 16 * 32; i += nthr) {
        const int r = i >> 5;
        const int c = (i & 31) * 4;
        const float4 xv = *(const float4*)(X + (rowBase + r) * DFEAT + c);
        As[r][c + 0] = xv.x; As[r][c + 1] = xv.y;
        As[r][c + 2] = xv.z; As[r][c + 3] = xv.w;
        if constexpr (FUSED) {
            const float4 gv = *(const float4*)(G + (rowBase + r) * DFEAT + c);
            Gs[r][c + 0] = gv.x; Gs[r][c + 1] = gv.y;
            Gs[r][c + 2] = gv.z; Gs[r][c + 3] = gv.w;
        }
    }
    __syncthreads();

    const int lane = tid & 31;
    const int wv   = tid >> 5;         // wave id = column tile
    const int m    = lane & 15;        // A row within tile / D column lane
    const int hi   = lane >> 4;        // half-wave select
    const int n    = wv * 16 + m;      // output column
    const int nc   = (n < NC) ? n : (NC - 1);  // clamp for masked tail tile

    // C/D layout: VGPR v <-> M = v + 8*hi, N = lane&15. Bias depends only on N.
    v8f acc;
    const float bb = bias[nc];
#pragma unroll
    for (int v = 0; v < 8; ++v) acc[v] = bb;

    // A frag: K = v + 2*hi (v in {0,1}) -> contiguous pair As[m][k], As[m][k+1]
    // B frag: same K mapping, N = lane&15.
#pragma unroll
    for (int kk = 0; kk < 32; ++kk) {
        const int k = kk * 4 + 2 * hi;
        const v2f a = *(const v2f*)&As[m][k];
        v2f b;
        b.x = W1[(long)k * NC + nc];
        b.y = W1[(long)(k + 1) * NC + nc];
        acc = __builtin_amdgcn_wmma_f32_16x16x4_f32(
            false, a, false, b, (short)0, acc, false, false);
    }
    if constexpr (FUSED) {
#pragma unroll
        for (int kk = 0; kk < 32; ++kk) {
            const int k = kk * 4 + 2 * hi;
            const v2f a = *(const v2f*)&Gs[m][k];
            v2f b;
            b.x = W2[(long)k * NC + nc];
            b.y = W2[(long)(k + 1) * NC + nc];
            acc = __builtin_amdgcn_wmma_f32_16x16x4_f32(
                false, a, false, b, (short)0, acc, false, false);
        }
    }

#pragma unroll
    for (int v = 0; v < 8; ++v) {
        float r = acc[v];
        if (RELU) r = fmaxf(r, 0.0f);
        if ((NC % 16 == 0) || (n < NC))
            out[(rowBase + v + 8 * hi) * NC + n] = r;
    }
}

// ---------------------------------------------------------------------------
// Zero-fill (agg init). Reference replaces -inf (no in-neighbors) with 0;
// with relu'd messages >= 0 a 0-init + max reproduces segment_max exactly.
// ---------------------------------------------------------------------------
__global__ void zero_f4(float4* __restrict__ p, long n4)
{
    const long i = (long)blockIdx.x * blockDim.x + threadIdx.x;
    if (i < n4) p[i] = make_float4(0.f, 0.f, 0.f, 0.f);
}

// ---------------------------------------------------------------------------
// Edge scatter-max: one wave per edge, 4 floats per lane (128 feats / 32).
// hp values are >= 0 (post-relu) so uint atomicMax == float max. Zero values
// are skipped (agg already 0) -> ~2x fewer atomics thanks to relu sparsity.
// ---------------------------------------------------------------------------
__global__ __launch_bounds__(256) void edge_scatter(
    const float* __restrict__ hp, const int* __restrict__ src,
    const int* __restrict__ dst, unsigned int* __restrict__ agg)
{
    const long gid  = (long)blockIdx.x * blockDim.x + threadIdx.x;
    const long e    = gid >> 5;
    const int  lane = (int)(gid & 31);
    if (e >= EDGES) return;

    const long s = src[e];
    const long d = dst[e];
    const float4 v = *(const float4*)(hp + s * DFEAT + lane * 4);
    unsigned int* base = agg + d * DFEAT + lane * 4;
    if (v.x > 0.0f) atomicMax(base + 0, __float_as_uint(v.x));
    if (v.y > 0.0f) atomicMax(base + 1, __float_as_uint(v.y));
    if (v.z > 0.0f) atomicMax(base + 2, __float_as_uint(v.z));
    if (v.w > 0.0f) atomicMax(base + 3, __float_as_uint(v.w));
}

// ---------------------------------------------------------------------------
extern "C" void kernel_launch(void* const* d_in, const int* in_sizes, int n_in,
                              void* d_out, int out_size, void* d_ws, size_t ws_size,
                              hipStream_t stream)
{
    (void)in_sizes; (void)n_in; (void)out_size; (void)ws_size;

    const float* X   = (const float*)d_in[0];
    const int*   src = (const int*)  d_in[1];
    const int*   dst = (const int*)  d_in[2];
    const float* Wp1 = (const float*)d_in[3];
    const float* bp1 = (const float*)d_in[4];
    const float* Ws1 = (const float*)d_in[5];
    const float* Wn1 = (const float*)d_in[6];
    const float* b1  = (const float*)d_in[7];
    const float* Wp2 = (const float*)d_in[8];
    const float* bp2 = (const float*)d_in[9];
    const float* Ws2 = (const float*)d_in[10];
    const float* Wn2 = (const float*)d_in[11];
    const float* b2  = (const float*)d_in[12];
    float*       out = (float*)d_out;

    const long NF  = (long)NODES * DFEAT;      // 12.8M floats = 51.2 MB
    float* hp  = (float*)d_ws;                 // [N,128] pooled features
    float* agg = hp + NF;                      // [N,128] segment-max result
    float* h   = agg + NF;                     // [N,128] layer-1 output

    const int  rowBlocks = NODES / 16;                         // 6250
    const long n4        = NF / 4;
    const int  zeroBlks  = (int)((n4 + 255) / 256);
    const int  edgeBlks  = (int)(((long)EDGES * 32 + 255) / 256);  // 200000

    // ---- Layer 1 ----
    sage_gemm<128, false, true><<<rowBlocks, 256, 0, stream>>>(
        X, nullptr, Wp1, nullptr, bp1, hp);                     // hp = relu(X@Wp1+bp1)
    zero_f4<<<zeroBlks, 256, 0, stream>>>((float4*)agg, n4);
    edge_scatter<<<edgeBlks, 256, 0, stream>>>(hp, src, dst, (unsigned int*)agg);
    sage_gemm<128, true, true><<<rowBlocks, 256, 0, stream>>>(
        X, agg, Ws1, Wn1, b1, h);                               // h = relu(X@Ws1 + agg@Wn1 + b1)

    // ---- Layer 2 ----
    sage_gemm<128, false, true><<<rowBlocks, 256, 0, stream>>>(
        h, nullptr, Wp2, nullptr, bp2, hp);                     // hp = relu(h@Wp2+bp2)
    zero_f4<<<zeroBlks, 256, 0, stream>>>((float4*)agg, n4);
    edge_scatter<<<edgeBlks, 256, 0, stream>>>(hp, src, dst, (unsigned int*)agg);
    sage_gemm<40, true, false><<<rowBlocks, 96, 0, stream>>>(
        h, agg, Ws2, Wn2, b2, out);                             // out = h@Ws2 + agg@Wn2 + b2
}